// GKTFM_81372450390826
// MI455X (gfx1250) — compile-verified
//
#include <hip/hip_runtime.h>
#include <math.h>

// ---------------------------------------------------------------------------
// GKT-FM on MI455X (gfx1250, wave32).
// Big GEMMs: V_WMMA_F32_16X16X4_F32, wave computes a 16x32 output tile
// (2 accumulators share one A fragment). K loops are split per input segment
// so the inner loop is branch-free (segment boundaries are multiples of 4,
// and ka = k0 + 2*hi never straddles one).
// GRU scan: 1 workgroup per batch row, Whh resident in 192KB LDS.
// ---------------------------------------------------------------------------

typedef float v2f __attribute__((ext_vector_type(2)));
typedef float v8f __attribute__((ext_vector_type(8)));

#define NQ   512
#define Hh   128
#define Bb   64
#define Ll   32
#define ROWS (Bb * NQ)          // 32768

__device__ __forceinline__ float sigmoidf_(float x) {
  return 1.0f / (1.0f + __expf(-x));
}

// Branch-free K-segment: acc{0,1} += A_seg x W[:, col0/col1].
// aBase: lane A pointer, already + 2*hi.  bBase*: W + (2*hi)*ldb + col.
template <int KSEG, int LDB>
__device__ __forceinline__ void gemm_seg(v8f& acc0, v8f& acc1,
                                         const float* __restrict__ aBase,
                                         const float* __restrict__ bBase0,
                                         const float* __restrict__ bBase1) {
#pragma unroll 8
  for (int k0 = 0; k0 < KSEG; k0 += 4) {
    v2f a = *(const v2f*)(aBase + k0);
    v2f b0, b1;
    b0.x = bBase0[k0 * LDB];
    b0.y = bBase0[k0 * LDB + LDB];
    b1.x = bBase1[k0 * LDB];
    b1.y = bBase1[k0 * LDB + LDB];
    acc0 = __builtin_amdgcn_wmma_f32_16x16x4_f32(false, a, false, b0,
                                                 (short)0, acc0, false, false);
    acc1 = __builtin_amdgcn_wmma_f32_16x16x4_f32(false, a, false, b1,
                                                 (short)0, acc1, false, false);
  }
}

// ---------------------------------------------------------------------------
// init: ht[b,n,:] = init_h[n,:]
// ---------------------------------------------------------------------------
__global__ void __launch_bounds__(256) k_init(const float* __restrict__ init_h,
                                              float* __restrict__ ht) {
  size_t i = (size_t)blockIdx.x * 256 + threadIdx.x;
  ht[i] = init_h[i & (size_t)(NQ * Hh - 1)];
}

// ---------------------------------------------------------------------------
// prep: tgt[b,0:128]=ht[b,qt,:], tgt[b,128:256]=x_emb[qt+512*rt,:]
//       Aij[b,n]=A[qt,n], Aji[b,n]=A[n,qt]
// ---------------------------------------------------------------------------
__global__ void __launch_bounds__(256) k_prep(const float* __restrict__ ht,
                                              const float* __restrict__ x_emb,
                                              const float* __restrict__ A,
                                              const int* __restrict__ q,
                                              const int* __restrict__ r,
                                              float* __restrict__ tgt,
                                              float* __restrict__ Aij,
                                              float* __restrict__ Aji, int t) {
  int b = blockIdx.x, j = threadIdx.x;
  int qt = q[b * Ll + t];
  int rt = r[b * Ll + t];
  int x  = qt + NQ * rt;
  if (j < 128) tgt[b * 256 + j] = ht[((size_t)(b * NQ + qt)) * Hh + j];
  else         tgt[b * 256 + j] = x_emb[(size_t)x * Hh + (j - 128)];
  for (int n = j; n < NQ; n += 256) {
    Aij[b * NQ + n] = A[(size_t)qt * NQ + n];
    Aji[b * NQ + n] = A[(size_t)n * NQ + qt];
  }
}

// ---------------------------------------------------------------------------
// neighbor layer 1 (WMMA): C1[row, 0:128]   = Aij[row]*relu(in @ Wo1 + bo1)
//                          C1[row, 128:256] = Aji[row]*relu(in @ Wi1 + bi1)
// in[row=(b,n), k] = k<256 ? tgt[b,k] : k<384 ? ht[row,k-256] : q_emb[n,k-384]
// M=32768, N=256, K=512. Wave: 16x32 tile. 16384 waves -> 2048 blocks.
// ---------------------------------------------------------------------------
__global__ void __launch_bounds__(256) k_neigh_l1(
    const float* __restrict__ tgt, const float* __restrict__ ht,
    const float* __restrict__ qemb,
    const float* __restrict__ Wo1, const float* __restrict__ Wi1,
    const float* __restrict__ bo1, const float* __restrict__ bi1,
    const float* __restrict__ Aij, const float* __restrict__ Aji,
    float* __restrict__ C1) {
  const int NPAIR = 256 / 32;                    // 8 tile-pairs per row tile
  int wave = (blockIdx.x * 256 + threadIdx.x) >> 5;
  int lane = threadIdx.x & 31;
  int tm = wave / NPAIR, tp = wave % NPAIR;
  int m0 = tm * 16, n0 = tp * 32;                // pair stays in one half
  int l = lane & 15, hi = lane >> 4;

  int row = m0 + l;
  int bidx = row >> 9, node = row & (NQ - 1);
  const float* aTg = tgt + bidx * 256 + 2 * hi;
  const float* aHt = ht + (size_t)row * Hh + 2 * hi;
  const float* aQe = qemb + (size_t)node * Hh + 2 * hi;

  const float* W1 = (n0 < 128) ? Wo1 : Wi1;
  int wc = n0 - ((n0 < 128) ? 0 : 128) + l;      // col inside Wo1/Wi1
  const float* b0 = W1 + (size_t)(2 * hi) * 128 + wc;
  const float* b1 = b0 + 16;

  v8f acc0 = {0.f, 0.f, 0.f, 0.f, 0.f, 0.f, 0.f, 0.f};
  v8f acc1 = acc0;
  gemm_seg<256, 128>(acc0, acc1, aTg, b0, b1);
  gemm_seg<128, 128>(acc0, acc1, aHt, b0 + 256 * 128, b1 + 256 * 128);
  gemm_seg<128, 128>(acc0, acc1, aQe, b0 + 384 * 128, b1 + 384 * 128);

  int cn0 = n0 + l, cn1 = cn0 + 16;
  const float* bvec = (n0 < 128) ? bo1 : bi1;
  const float* scl  = (n0 < 128) ? Aij : Aji;
  float bias0 = bvec[cn0 - ((n0 < 128) ? 0 : 128)];
  float bias1 = bvec[cn1 - ((n0 < 128) ? 0 : 128)];
#pragma unroll
  for (int rr = 0; rr < 8; ++rr) {
    int rm = m0 + rr + 8 * hi;
    float s = scl[rm];
    float v0 = acc0[rr] + bias0;
    float v1 = acc1[rr] + bias1;
    C1[(size_t)rm * 256 + cn0] = (v0 > 0.f ? v0 : 0.f) * s;
    C1[(size_t)rm * 256 + cn1] = (v1 > 0.f ? v1 : 0.f) * s;
  }
}

// ---------------------------------------------------------------------------
// neighbor layer 2 (WMMA): m[row,:] = C1[row,0:128]@Wo2 + C1[row,128:256]@Wi2
//                                     + Aij[row]*bo2 + Aji[row]*bi2
// M=32768, N=128, K=256. 8192 waves -> 1024 blocks.
// ---------------------------------------------------------------------------
__global__ void __launch_bounds__(256) k_neigh_l2(
    const float* __restrict__ C1,
    const float* __restrict__ Wo2, const float* __restrict__ Wi2,
    const float* __restrict__ bo2, const float* __restrict__ bi2,
    const float* __restrict__ Aij, const float* __restrict__ Aji,
    float* __restrict__ mout) {
  const int NPAIR = 128 / 32;                    // 4
  int wave = (blockIdx.x * 256 + threadIdx.x) >> 5;
  int lane = threadIdx.x & 31;
  int tm = wave / NPAIR, tp = wave % NPAIR;
  int m0 = tm * 16, n0 = tp * 32;
  int l = lane & 15, hi = lane >> 4;

  int row = m0 + l;
  const float* aC = C1 + (size_t)row * 256 + 2 * hi;
  int nc0 = n0 + l;
  const float* bo = Wo2 + (size_t)(2 * hi) * 128 + nc0;
  const float* bi = Wi2 + (size_t)(2 * hi) * 128 + nc0;

  v8f acc0 = {0.f, 0.f, 0.f, 0.f, 0.f, 0.f, 0.f, 0.f};
  v8f acc1 = acc0;
  gemm_seg<128, 128>(acc0, acc1, aC, bo, bo + 16);
  gemm_seg<128, 128>(acc0, acc1, aC + 128, bi, bi + 16);

  int nc1 = nc0 + 16;
#pragma unroll
  for (int rr = 0; rr < 8; ++rr) {
    int rm = m0 + rr + 8 * hi;
    float ai = Aij[rm], aj = Aji[rm];
    mout[(size_t)rm * 128 + nc0] = acc0[rr] + ai * bo2[nc0] + aj * bi2[nc0];
    mout[(size_t)rm * 128 + nc1] = acc1[rr] + ai * bo2[nc1] + aj * bi2[nc1];
  }
}

// ---------------------------------------------------------------------------
// self MLP at answered node only (64 rows): m[b,qt,:] = MLP_self(tgt[b])
// ---------------------------------------------------------------------------
__global__ void __launch_bounds__(128) k_self(
    const float* __restrict__ tgt,
    const float* __restrict__ Ws1, const float* __restrict__ bs1,
    const float* __restrict__ Ws2, const float* __restrict__ bs2,
    const int* __restrict__ q, float* __restrict__ mout, int t) {
  __shared__ float st[256];
  __shared__ float sh1[128];
  int b = blockIdx.x, j = threadIdx.x;
  st[j]       = tgt[b * 256 + j];
  st[j + 128] = tgt[b * 256 + 128 + j];
  __syncthreads();
  float a = bs1[j];
#pragma unroll 8
  for (int k = 0; k < 256; ++k) a += st[k] * Ws1[(size_t)k * 128 + j];
  sh1[j] = fmaxf(a, 0.f);
  __syncthreads();
  float o = bs2[j];
#pragma unroll 8
  for (int k = 0; k < 128; ++k) o += sh1[k] * Ws2[(size_t)k * 128 + j];
  int qt = q[b * Ll + t];
  mout[((size_t)(b * NQ + qt)) * 128 + j] = o;
}

// ---------------------------------------------------------------------------
// gx (WMMA): gx[row,:] = [m[row,:], ht[row,:]] @ Wih + bih
// M=32768, N=384, K=256. 24576 waves -> 3072 blocks.
// ---------------------------------------------------------------------------
__global__ void __launch_bounds__(256) k_gx(
    const float* __restrict__ mbuf, const float* __restrict__ ht,
    const float* __restrict__ Wih, const float* __restrict__ bih,
    float* __restrict__ gx) {
  const int NPAIR = 384 / 32;                    // 12
  int wave = (blockIdx.x * 256 + threadIdx.x) >> 5;
  int lane = threadIdx.x & 31;
  int tm = wave / NPAIR, tp = wave % NPAIR;
  int m0 = tm * 16, n0 = tp * 32;
  int l = lane & 15, hi = lane >> 4;

  int row = m0 + l;
  const float* aM = mbuf + (size_t)row * 128 + 2 * hi;
  const float* aH = ht + (size_t)row * 128 + 2 * hi;
  int nc0 = n0 + l;
  const float* b0 = Wih + (size_t)(2 * hi) * 384 + nc0;
  const float* b1 = b0 + 16;

  v8f acc0 = {0.f, 0.f, 0.f, 0.f, 0.f, 0.f, 0.f, 0.f};
  v8f acc1 = acc0;
  gemm_seg<128, 384>(acc0, acc1, aM, b0, b1);
  gemm_seg<128, 384>(acc0, acc1, aH, b0 + 128 * 384, b1 + 128 * 384);

  int nc1 = nc0 + 16;
#pragma unroll
  for (int rr = 0; rr < 8; ++rr) {
    int rm = m0 + rr + 8 * hi;
    gx[(size_t)rm * 384 + nc0] = acc0[rr] + bih[nc0];
    gx[(size_t)rm * 384 + nc1] = acc1[rr] + bih[nc1];
  }
}

// ---------------------------------------------------------------------------
// GRU scan over nodes. 1 block / batch row; Whh (192KB) in dynamic LDS.
// ---------------------------------------------------------------------------
__global__ void __launch_bounds__(384) k_gru_scan(
    const float* __restrict__ gx, const float* __restrict__ Whh,
    const float* __restrict__ bhh, float* __restrict__ htn) {
  extern __shared__ float sW[];                  // [128][384]
  __shared__ float sh[128];
  __shared__ float sg[384];
  int b = blockIdx.x, tid = threadIdx.x;
  for (int i = tid; i < 128 * 384; i += 384) sW[i] = Whh[i];
  if (tid < 128) sh[tid] = 0.f;
  float bb = bhh[tid];
  __syncthreads();

  for (int n = 0; n < NQ; ++n) {
    const float* g = gx + ((size_t)(b * NQ + n)) * 384;
    float xr = 0.f, xz = 0.f, xn = 0.f;
    if (tid < 128) {                             // issue early, used after dot
      xr = g[tid];
      xz = g[tid + 128];
      xn = g[tid + 256];
    }
    float acc = bb;
#pragma unroll 8
    for (int k = 0; k < 128; ++k) acc += sh[k] * sW[k * 384 + tid];
    sg[tid] = acc;
    __syncthreads();
    if (tid < 128) {
      float hr = sg[tid], hz = sg[tid + 128], hn = sg[tid + 256];
      float rg = sigmoidf_(xr + hr);
      float zg = sigmoidf_(xz + hz);
      float ng = tanhf(xn + rg * hn);
      float h  = (1.f - zg) * ng + zg * sh[tid];
      sh[tid] = h;
      htn[((size_t)(b * NQ + n)) * 128 + tid] = h;
    }
    __syncthreads();
  }
}

// ---------------------------------------------------------------------------
// logits for all nodes: out[b,t,n] = sigmoid(ht_new[b,n,:].outW + bias[n])
// ---------------------------------------------------------------------------
__global__ void __launch_bounds__(256) k_logits(
    const float* __restrict__ htn, const float* __restrict__ outW,
    const float* __restrict__ bias, float* __restrict__ out, int t) {
  int wave = (blockIdx.x * 256 + threadIdx.x) >> 5;
  int lane = threadIdx.x & 31;
  int b = wave >> 9, n = wave & (NQ - 1);
  const float4 hv = *(const float4*)(htn + (size_t)wave * 128 + lane * 4);
  const float4 wv = *(const float4*)(outW + lane * 4);
  float s = hv.x * wv.x + hv.y * wv.y + hv.z * wv.z + hv.w * wv.w;
#pragma unroll
  for (int off = 16; off > 0; off >>= 1) s += __shfl_xor(s, off, 32);
  if (lane == 0)
    out[((size_t)b * Ll + t) * NQ + n] = sigmoidf_(s + bias[n]);
}

// ---------------------------------------------------------------------------
// flow-matching at answered node + overwrite out[b,t,qt].
// ---------------------------------------------------------------------------
__global__ void __launch_bounds__(128) k_fm(
    const float* __restrict__ htn, const int* __restrict__ q,
    const float* __restrict__ Wt1, const float* __restrict__ bt1,
    const float* __restrict__ Wt2, const float* __restrict__ bt2,
    const float* __restrict__ W1, const float* __restrict__ b1,
    const float* __restrict__ W2, const float* __restrict__ b2,
    const float* __restrict__ W3, const float* __restrict__ b3,
    const float* __restrict__ outW, const float* __restrict__ bias,
    float* __restrict__ out, int t) {
  __shared__ float sx[160];                      // [xhat(128) | te(32)]
  __shared__ float s1[128];
  __shared__ float s2[128];
  __shared__ float ste1[32];
  int b = blockIdx.x, j = threadIdx.x;
  int qt = q[b * Ll + t];
  sx[j] = htn[((size_t)(b * NQ + qt)) * 128 + j];
  __syncthreads();

  for (int s = 0; s < 4; ++s) {
    float tval = (s + 0.5f) * 0.25f;
    if (j < 32) ste1[j] = fmaxf(tval * Wt1[j] + bt1[j], 0.f);
    __syncthreads();
    if (j < 32) {
      float a = bt2[j];
#pragma unroll
      for (int k = 0; k < 32; ++k) a += ste1[k] * Wt2[k * 32 + j];
      sx[128 + j] = fmaxf(a, 0.f);
    }
    __syncthreads();
    float a = b1[j];
#pragma unroll 8
    for (int k = 0; k < 160; ++k) a += sx[k] * W1[(size_t)k * 128 + j];
    s1[j] = fmaxf(a, 0.f);
    __syncthreads();
    a = b2[j];
#pragma unroll 8
    for (int k = 0; k < 128; ++k) a += s1[k] * W2[(size_t)k * 128 + j];
    s2[j] = fmaxf(a, 0.f);
    __syncthreads();
    a = b3[j];
#pragma unroll 8
    for (int k = 0; k < 128; ++k) a += s2[k] * W3[(size_t)k * 128 + j];
    __syncthreads();
    sx[j] += 0.25f * a;                          // xhat += dt * v
    __syncthreads();
  }
  s1[j] = sx[j] * outW[j];
  __syncthreads();
#pragma unroll
  for (int off = 64; off > 0; off >>= 1) {
    if (j < off) s1[j] += s1[j + off];
    __syncthreads();
  }
  if (j == 0)
    out[((size_t)b * Ll + t) * NQ + qt] = sigmoidf_(s1[0] + bias[qt]);
}

// ---------------------------------------------------------------------------
// host orchestration
// ---------------------------------------------------------------------------
extern "C" void kernel_launch(void* const* d_in, const int* in_sizes, int n_in,
                              void* d_out, int out_size, void* d_ws,
                              size_t ws_size, hipStream_t stream) {
  (void)in_sizes; (void)n_in; (void)out_size; (void)ws_size;

  const int*   q      = (const int*)d_in[0];
  const int*   r      = (const int*)d_in[1];
  const float* x_emb  = (const float*)d_in[2];
  const float* q_emb  = (const float*)d_in[3];
  const float* init_h = (const float*)d_in[4];
  const float* A      = (const float*)d_in[5];
  const float* Ws1 = (const float*)d_in[6],  *bs1 = (const float*)d_in[7];
  const float* Ws2 = (const float*)d_in[8],  *bs2 = (const float*)d_in[9];
  const float* Wo1 = (const float*)d_in[10], *bo1 = (const float*)d_in[11];
  const float* Wo2 = (const float*)d_in[12], *bo2 = (const float*)d_in[13];
  const float* Wi1 = (const float*)d_in[14], *bi1 = (const float*)d_in[15];
  const float* Wi2 = (const float*)d_in[16], *bi2 = (const float*)d_in[17];
  const float* Wih = (const float*)d_in[18];
  const float* Whh = (const float*)d_in[19];
  const float* bih = (const float*)d_in[20];
  const float* bhh = (const float*)d_in[21];
  const float* Wt1 = (const float*)d_in[22], *bt1 = (const float*)d_in[23];
  const float* Wt2 = (const float*)d_in[24], *bt2 = (const float*)d_in[25];
  const float* vW1 = (const float*)d_in[26], *vb1 = (const float*)d_in[27];
  const float* vW2 = (const float*)d_in[28], *vb2 = (const float*)d_in[29];
  const float* vW3 = (const float*)d_in[30], *vb3 = (const float*)d_in[31];
  const float* outW = (const float*)d_in[32];
  const float* bias = (const float*)d_in[33];
  float* out = (float*)d_out;

  float* ws = (float*)d_ws;
  size_t off = 0;
  auto alloc = [&](size_t nfloats) {
    float* p = ws + off;
    off += (nfloats + 63) & ~(size_t)63;        // 256B-align each region
    return p;
  };
  float* htA  = alloc((size_t)ROWS * Hh);       // 16MB
  float* htB  = alloc((size_t)ROWS * Hh);       // 16MB
  float* mbuf = alloc((size_t)ROWS * Hh);       // 16MB
  float* gx   = alloc((size_t)ROWS * 384);      // 48MB  (C1 aliases this)
  float* C1   = gx;                             // C1 [ROWS,256] dead before gx
  float* tgt  = alloc((size_t)Bb * 256);
  float* Aij  = alloc((size_t)Bb * NQ);
  float* Aji  = alloc((size_t)Bb * NQ);

  k_init<<<(ROWS * Hh) / 256, 256, 0, stream>>>(init_h, htA);

  float* ht  = htA;
  float* htn = htB;
  for (int t = 0; t < Ll; ++t) {
    k_prep<<<Bb, 256, 0, stream>>>(ht, x_emb, A, q, r, tgt, Aij, Aji, t);
    // (32768/16)*(256/32) = 16384 waves -> 2048 blocks
    k_neigh_l1<<<2048, 256, 0, stream>>>(tgt, ht, q_emb, Wo1, Wi1, bo1, bi1,
                                         Aij, Aji, C1);
    // (32768/16)*(128/32) = 8192 waves -> 1024 blocks
    k_neigh_l2<<<1024, 256, 0, stream>>>(C1, Wo2, Wi2, bo2, bi2, Aij, Aji,
                                         mbuf);
    k_self<<<Bb, 128, 0, stream>>>(tgt, Ws1, bs1, Ws2, bs2, q, mbuf, t);
    // (32768/16)*(384/32) = 24576 waves -> 3072 blocks
    k_gx<<<3072, 256, 0, stream>>>(mbuf, ht, Wih, bih, gx);
    k_gru_scan<<<Bb, 384, 128 * 384 * sizeof(float), stream>>>(gx, Whh, bhh,
                                                               htn);
    // 32768 waves -> 4096 blocks
    k_logits<<<4096, 256, 0, stream>>>(htn, outW, bias, out, t);
    k_fm<<<Bb, 128, 0, stream>>>(htn, q, Wt1, bt1, Wt2, bt2, vW1, vb1, vW2,
                                 vb2, vW3, vb3, outW, bias, out, t);
    float* tmp = ht; ht = htn; htn = tmp;
  }
}